// BindingAffinityModel_29454885716582
// MI455X (gfx1250) — compile-verified
//
#include <hip/hip_runtime.h>
#include <hip/hip_bf16.h>

typedef __attribute__((ext_vector_type(2))) float v2f;
typedef __attribute__((ext_vector_type(8))) float v8f;

// ---------------------------------------------------------------------------
// Generic fp32 WMMA GEMM: C[M,N] = A[M,K] @ B[K,N]
// grid.x = M/16 tiles, grid.y = N/16 tiles, one wave (32 threads) per tile.
// Uses V_WMMA_F32_16X16X4_F32; K must be a multiple of 4, M/N multiples of 16.
// Lane layouts per CDNA5 ISA 7.12.2:
//   A 16x4 : lanes 0-15 rows M=0..15 hold {K=k,k+1}; lanes 16-31 hold {K=k+2,k+3}
//   B 4x16 : vgpr0: lanes0-15 K=k, lanes16-31 K=k+2; vgpr1: K=k+1 / K=k+3
//   C 16x16: vgpr r: lanes0-15 -> M=r, lanes16-31 -> M=r+8; N = lane&15
// ---------------------------------------------------------------------------
__global__ __launch_bounds__(32) void wmma_gemm_f32(const float* __restrict__ A,
                                                    const float* __restrict__ B,
                                                    float* __restrict__ C,
                                                    int K, int N) {
  const int lane = threadIdx.x;
  const int half = lane >> 4;   // 0 or 1
  const int l16  = lane & 15;
  const int row0 = blockIdx.x * 16;
  const int col0 = blockIdx.y * 16;

  const float* arow = A + (size_t)(row0 + l16) * K;
  v8f acc = {};
  for (int k = 0; k < K; k += 4) {
    v2f a, b;
    a.x = arow[k + 2 * half + 0];
    a.y = arow[k + 2 * half + 1];
    b.x = B[(size_t)(k + 2 * half + 0) * N + col0 + l16];
    b.y = B[(size_t)(k + 2 * half + 1) * N + col0 + l16];
    acc = __builtin_amdgcn_wmma_f32_16x16x4_f32(false, a, false, b,
                                                (short)0, acc, false, false);
  }
  float* crow = C + (size_t)(row0 + 8 * half) * N + col0 + l16;
#pragma unroll
  for (int r = 0; r < 8; ++r) crow[(size_t)r * N] = acc[r];
}

// ---------------------------------------------------------------------------
// Small helper kernels
// ---------------------------------------------------------------------------
__global__ void zero_f32(float* __restrict__ p, int n) {
  int i = blockIdx.x * blockDim.x + threadIdx.x;
  if (i < n) p[i] = 0.0f;
}

// deg[dst] += 1 for every edge (in-degree)
__global__ void deg_count(const int* __restrict__ dst, float* __restrict__ deg, int E) {
  int e = blockIdx.x * blockDim.x + threadIdx.x;
  if (e < E) atomicAdd(&deg[dst[e]], 1.0f);
}

// dinv[i] = rsqrt(deg[i] + 1)   (self-loop included)
__global__ void deg_finalize(float* __restrict__ dinv, int n) {
  int i = blockIdx.x * blockDim.x + threadIdx.x;
  if (i < n) dinv[i] = rsqrtf(dinv[i] + 1.0f);
}

// Layer-1 "GEMM": H[i,j] = x[i] * W1[j]  (K = 1, outer product)
__global__ void l1_gemm(const float* __restrict__ x, const float* __restrict__ W,
                        float* __restrict__ H, int total /* = N*64 */) {
  int i = blockIdx.x * blockDim.x + threadIdx.x;
  if (i < total) H[i] = x[i >> 6] * W[i & 63];
}

// agg[i,:] = h[i,:] * dinv[i]^2  (self-loop contribution, initializes agg)
__global__ void selfloop_init(const float* __restrict__ h, const float* __restrict__ dinv,
                              float* __restrict__ agg, int total, int shift /* log2(D) */) {
  int i = blockIdx.x * blockDim.x + threadIdx.x;
  if (i < total) {
    float dv = dinv[i >> shift];
    agg[i] = h[i] * (dv * dv);
  }
}

// Per (edge, 4-feature chunk): agg[dst,:] += h[src,:] * dinv[src]*dinv[dst]
// cshift = log2(D/4); row stride D = 4 << cshift
__global__ void edge_scatter(const int* __restrict__ src, const int* __restrict__ dst,
                             const float* __restrict__ h, const float* __restrict__ dinv,
                             float* __restrict__ agg, int E, int cshift) {
  int idx = blockIdx.x * blockDim.x + threadIdx.x;
  int e = idx >> cshift;
  if (e >= E) return;
  int q = idx & ((1 << cshift) - 1);
  int s = src[e];
  int d = dst[e];
  float c = dinv[s] * dinv[d];
  const float4 hv = *(const float4*)(h + ((size_t)s << (cshift + 2)) + q * 4);
  float* ap = agg + ((size_t)d << (cshift + 2)) + q * 4;
  atomicAdd(ap + 0, hv.x * c);
  atomicAdd(ap + 1, hv.y * c);
  atomicAdd(ap + 2, hv.z * c);
  atomicAdd(ap + 3, hv.w * c);
}

// agg[i,j] = relu(agg[i,j] + b[j])   (mask = D-1)
__global__ void bias_relu(float* __restrict__ agg, const float* __restrict__ b,
                          int total, int mask) {
  int i = blockIdx.x * blockDim.x + threadIdx.x;
  if (i < total) {
    float v = agg[i] + b[i & mask];
    agg[i] = v > 0.0f ? v : 0.0f;
  }
}

// pooled[batch[node], j] += h[node, j]; cnts[batch[node]] += 1 (once per node)
__global__ void pool_scatter(const float* __restrict__ h, const int* __restrict__ batch,
                             float* __restrict__ pooled, float* __restrict__ cnts, int total) {
  int idx = blockIdx.x * blockDim.x + threadIdx.x;
  if (idx >= total) return;
  int node = idx >> 6;
  int j = idx & 63;
  int g = batch[node];
  atomicAdd(&pooled[g * 64 + j], h[idx]);
  if (j == 0) atomicAdd(&cnts[g], 1.0f);
}

// pooled[g,j] /= max(cnts[g], 1)
__global__ void pool_div(float* __restrict__ pooled, const float* __restrict__ cnts, int total) {
  int idx = blockIdx.x * blockDim.x + threadIdx.x;
  if (idx < total) pooled[idx] /= fmaxf(cnts[idx >> 6], 1.0f);
}

// out[g] = sum_j relu(z[g,j] + lin1_b[j]) * lin2_w[j] + lin2_b[0]
// one wave (32 lanes == 32 hidden units) per graph, 8 waves per block
__global__ __launch_bounds__(256) void head(const float* __restrict__ z,
                                            const float* __restrict__ lin1_b,
                                            const float* __restrict__ lin2_w,
                                            const float* __restrict__ lin2_b,
                                            float* __restrict__ out, int G) {
  int g = blockIdx.x * 8 + (threadIdx.x >> 5);
  int lane = threadIdx.x & 31;
  if (g >= G) return;
  float v = fmaxf(z[g * 32 + lane] + lin1_b[lane], 0.0f) * lin2_w[lane];
#pragma unroll
  for (int off = 16; off > 0; off >>= 1) v += __shfl_xor(v, off, 32);
  if (lane == 0) out[g] = v + lin2_b[0];
}

// ---------------------------------------------------------------------------
// Launch
// ---------------------------------------------------------------------------
extern "C" void kernel_launch(void* const* d_in, const int* in_sizes, int n_in,
                              void* d_out, int out_size, void* d_ws, size_t ws_size,
                              hipStream_t stream) {
  const float* x      = (const float*)d_in[0];
  const float* W1     = (const float*)d_in[1];
  const float* b1     = (const float*)d_in[2];
  const float* W2     = (const float*)d_in[3];
  const float* b2     = (const float*)d_in[4];
  const float* W3     = (const float*)d_in[5];
  const float* b3     = (const float*)d_in[6];
  const float* lin1_w = (const float*)d_in[7];
  const float* lin1_b = (const float*)d_in[8];
  const float* lin2_w = (const float*)d_in[9];
  const float* lin2_b = (const float*)d_in[10];
  const int*   edge   = (const int*)d_in[11];   // [2,E] row-major
  const int*   batch  = (const int*)d_in[12];

  const int N = in_sizes[0];        // 50000 (multiple of 16)
  const int E = in_sizes[11] / 2;   // 600000
  const int G = out_size;           // 256
  const int* esrc = edge;
  const int* edst = edge + E;

  float* ws   = (float*)d_ws;
  float* dinv = ws;                       // [N]
  float* hA   = ws + N;                   // [N,128] max
  float* hB   = hA + (size_t)N * 128;     // [N,128] max
  float* pooled = hA;                     // [G,64]  (hA free after last conv)
  float* cnts   = pooled + G * 64;        // [G]
  float* z      = cnts + G;               // [G,32]

  const int T = 256;
  auto blocks = [](int n, int t) { return (n + t - 1) / t; };

  // --- degree -> dinv (shared by all three convs) ---
  zero_f32<<<blocks(N, T), T, 0, stream>>>(dinv, N);
  deg_count<<<blocks(E, T), T, 0, stream>>>(edst, dinv, E);
  deg_finalize<<<blocks(N, T), T, 0, stream>>>(dinv, N);

  // --- conv1: x[N,1] @ W1[1,64] -> 64 ---
  l1_gemm<<<blocks(N * 64, T), T, 0, stream>>>(x, W1, hA, N * 64);
  selfloop_init<<<blocks(N * 64, T), T, 0, stream>>>(hA, dinv, hB, N * 64, 6);
  edge_scatter<<<blocks(E * 16, T), T, 0, stream>>>(esrc, edst, hA, dinv, hB, E, 4);
  bias_relu<<<blocks(N * 64, T), T, 0, stream>>>(hB, b1, N * 64, 63);

  // --- conv2: hB[N,64] @ W2[64,128] -> 128 ---
  wmma_gemm_f32<<<dim3(N / 16, 128 / 16), 32, 0, stream>>>(hB, W2, hA, 64, 128);
  selfloop_init<<<blocks(N * 128, T), T, 0, stream>>>(hA, dinv, hB, N * 128, 7);
  edge_scatter<<<blocks(E * 32, T), T, 0, stream>>>(esrc, edst, hA, dinv, hB, E, 5);
  bias_relu<<<blocks(N * 128, T), T, 0, stream>>>(hB, b2, N * 128, 127);

  // --- conv3: hB[N,128] @ W3[128,64] -> 64 ---
  wmma_gemm_f32<<<dim3(N / 16, 64 / 16), 32, 0, stream>>>(hB, W3, hA, 128, 64);
  selfloop_init<<<blocks(N * 64, T), T, 0, stream>>>(hA, dinv, hB, N * 64, 6);
  edge_scatter<<<blocks(E * 16, T), T, 0, stream>>>(esrc, edst, hA, dinv, hB, E, 4);
  bias_relu<<<blocks(N * 64, T), T, 0, stream>>>(hB, b3, N * 64, 63);

  // --- global mean pool (hA region reused for pooled/cnts/z) ---
  zero_f32<<<blocks(G * 64 + G + G * 32, T), T, 0, stream>>>(pooled, G * 64 + G + G * 32);
  pool_scatter<<<blocks(N * 64, T), T, 0, stream>>>(hB, batch, pooled, cnts, N * 64);
  pool_div<<<blocks(G * 64, T), T, 0, stream>>>(pooled, cnts, G * 64);

  // --- head: pooled[G,64] @ lin1_w[64,32] -> z, then relu+dot with lin2 ---
  wmma_gemm_f32<<<dim3(G / 16, 32 / 16), 32, 0, stream>>>(pooled, lin1_w, z, 64, 32);
  head<<<blocks(G * 32, 256), 256, 0, stream>>>(z, lin1_b, lin2_w, lin2_b, (float*)d_out, G);
}